// LightGCN_39943195853183
// MI455X (gfx1250) — compile-verified
//
#include <hip/hip_runtime.h>
#include <stdint.h>

namespace {
constexpr int       kUsers = 100000;
constexpr int       kItems = 50000;
constexpr int       kN     = kUsers + kItems;     // 150000 nodes
constexpr int       kD     = 64;                  // latent dim
constexpr int       kNNZ   = 2400000;             // edges
constexpr long long kNE    = (long long)kN * kD;  // 9,600,000 floats per buffer
}

// ---------------------------------------------------------------------------
// 1) concat(user_emb, item_emb) -> cur, and acc = cur (layer-0 snapshot)
// ---------------------------------------------------------------------------
__global__ __launch_bounds__(256)
void init_concat(const float* __restrict__ user_emb,
                 const float* __restrict__ item_emb,
                 float* __restrict__ cur, float* __restrict__ acc) {
  const long long i = (long long)blockIdx.x * blockDim.x + threadIdx.x; // float4 idx
  const long long u4 = (long long)kUsers * kD / 4;
  if (i >= kNE / 4) return;
  const float4 v = (i < u4) ? ((const float4*)user_emb)[i]
                            : ((const float4*)item_emb)[i - u4];
  ((float4*)cur)[i] = v;
  ((float4*)acc)[i] = v;
}

// ---------------------------------------------------------------------------
// 2) CSR construction: zero counts, histogram rows, scan, scatter
// ---------------------------------------------------------------------------
__global__ __launch_bounds__(256)
void zero_i32(int* __restrict__ p, int n) {
  int i = blockIdx.x * blockDim.x + threadIdx.x;
  if (i < n) p[i] = 0;
}

__global__ __launch_bounds__(256)
void hist_rows(const int* __restrict__ rows, int* __restrict__ counts) {
  int e = blockIdx.x * blockDim.x + threadIdx.x;
  if (e < kNNZ) atomicAdd(&counts[rows[e]], 1);
}

// Single-workgroup exclusive scan over kN counters (147 chunks of 1024).
__global__ __launch_bounds__(1024)
void scan_counts(const int* __restrict__ counts, int* __restrict__ row_ptr,
                 int* __restrict__ row_fill, int n) {
  __shared__ int s[1024];
  __shared__ int carry_s;
  const int tid = threadIdx.x;
  if (tid == 0) carry_s = 0;
  __syncthreads();
  for (int base = 0; base < n; base += 1024) {
    const int i = base + tid;
    const int x = (i < n) ? counts[i] : 0;
    s[tid] = x;
    __syncthreads();
    for (int off = 1; off < 1024; off <<= 1) {      // Hillis–Steele inclusive
      int t = (tid >= off) ? s[tid - off] : 0;
      __syncthreads();
      s[tid] += t;
      __syncthreads();
    }
    const int excl  = s[tid] - x;
    const int carry = carry_s;
    if (i < n) { row_ptr[i] = carry + excl; row_fill[i] = carry + excl; }
    __syncthreads();
    if (tid == 0) carry_s += s[1023];
    __syncthreads();
  }
  if (tid == 0) row_ptr[n] = carry_s;
}

// Scatter edges into CSR order as interleaved (col, val) pairs so the SpMM
// can stage a whole chunk with a single async b64 per lane.
__global__ __launch_bounds__(256)
void scatter_edges(const int* __restrict__ rows, const int* __restrict__ cols,
                   const float* __restrict__ vals, int* __restrict__ row_fill,
                   int2* __restrict__ edge_pairs) {
  int e = blockIdx.x * blockDim.x + threadIdx.x;
  if (e >= kNNZ) return;
  int p = atomicAdd(&row_fill[rows[e]], 1);
  edge_pairs[p] = make_int2(cols[e], __float_as_int(vals[e]));
}

// ---------------------------------------------------------------------------
// 3) Fused pull-SpMM + L2-normalize + layer-mean accumulate.
//    One wave32 per output row; lane l owns dims {2l, 2l+1}.
//    Edge-list chunks are staged into LDS via global_load_async_to_lds_b64
//    (ASYNCcnt); gather rows are prefetched one edge ahead into L0/L2.
// ---------------------------------------------------------------------------
__global__ __launch_bounds__(256)
void spmm_norm_acc(const float* __restrict__ cur, float* __restrict__ nxt,
                   float* __restrict__ acc, const int* __restrict__ row_ptr,
                   const int2* __restrict__ edge_pairs) {
  __shared__ int2 s_edge[8][32];
  const int lane = threadIdx.x & 31;
  const int wid  = threadIdx.x >> 5;
  const int row  = blockIdx.x * 8 + wid;
  if (row >= kN) return;

  const int beg = row_ptr[row];
  const int end = row_ptr[row + 1];
  float ax = 0.0f, ay = 0.0f;

#if defined(__gfx1250__)
  // Workgroup-relative LDS byte offset (flat shared addr truncated to 32 bits).
  const uint32_t lds_e = (uint32_t)(uintptr_t)&s_edge[wid][lane];
#endif

  for (int base = beg; base < end; base += 32) {
    const int cnt = min(32, end - base);
    // Prefetch next chunk's edge stream while we work on this one.
    if (base + 32 < end) __builtin_prefetch(&edge_pairs[base + 32], 0, 0);
#if defined(__gfx1250__)
    // Retire prior-chunk LDS reads before the async engine overwrites the
    // staging buffer, then issue one async b64 (col,val) per active lane.
    asm volatile("s_wait_dscnt 0x0" ::: "memory");
    if (lane < cnt) {
      const uint64_t ge = (uint64_t)(uintptr_t)(edge_pairs + base + lane);
      asm volatile("global_load_async_to_lds_b64 %0, %1, off"
                   :: "v"(lds_e), "v"(ge) : "memory");
    }
    asm volatile("s_wait_asynccnt 0x0" ::: "memory");
#else
    if (lane < cnt) s_edge[wid][lane] = edge_pairs[base + lane];
    __builtin_amdgcn_wave_barrier();
#endif
    for (int j = 0; j < cnt; ++j) {
      const int2  ev = s_edge[wid][j];            // uniform LDS b64 broadcast
      const int   c  = ev.x;
      const float v  = __int_as_float(ev.y);
      if (j + 1 < cnt) {                          // hide L2 gather latency
        const int cn = s_edge[wid][j + 1].x;
        __builtin_prefetch(&cur[(size_t)cn * kD + 2 * lane], 0, 0);
      }
      const float2 f = *(const float2*)&cur[(size_t)c * kD + 2 * lane];
      ax = fmaf(v, f.x, ax);
      ay = fmaf(v, f.y, ay);
    }
  }

  // Row L2 norm: wave32 butterfly reduction of sum of squares.
  float ss = ax * ax + ay * ay;
  for (int off = 16; off > 0; off >>= 1) ss += __shfl_xor(ss, off, 32);
  const float scale = 1.0f / fmaxf(sqrtf(ss), 1e-12f);
  const float nx = ax * scale, ny = ay * scale;

  const size_t o = (size_t)row * kD + 2 * lane;
  *(float2*)&nxt[o] = make_float2(nx, ny);
  float2 a = *(float2*)&acc[o];
  a.x += nx; a.y += ny;
  *(float2*)&acc[o] = a;
}

// ---------------------------------------------------------------------------
// 4) mean over 4 snapshots: out = acc * 0.25
// ---------------------------------------------------------------------------
__global__ __launch_bounds__(256)
void finalize_mean(const float* __restrict__ acc, float* __restrict__ out) {
  const long long i = (long long)blockIdx.x * blockDim.x + threadIdx.x;
  if (i >= kNE / 4) return;
  float4 a = ((const float4*)acc)[i];
  a.x *= 0.25f; a.y *= 0.25f; a.z *= 0.25f; a.w *= 0.25f;
  ((float4*)out)[i] = a;
}

// ---------------------------------------------------------------------------
extern "C" void kernel_launch(void* const* d_in, const int* in_sizes, int n_in,
                              void* d_out, int out_size, void* d_ws,
                              size_t ws_size, hipStream_t stream) {
  (void)in_sizes; (void)n_in; (void)out_size; (void)ws_size;
  const float* user_emb = (const float*)d_in[0];
  const float* item_emb = (const float*)d_in[1];
  const float* vals     = (const float*)d_in[2];
  const int*   rows     = (const int*)d_in[3];
  const int*   cols     = (const int*)d_in[4];
  float*       out      = (float*)d_out;

  // Workspace carve-out (~136 MB): cur | nxt | acc | counts | row_ptr(+pad) |
  // row_fill | edge_pairs   (edge_pairs kept 8B-aligned)
  float* cur      = (float*)d_ws;
  float* nxt      = cur + kNE;
  float* acc      = nxt + kNE;
  int*   counts   = (int*)(acc + kNE);
  int*   row_ptr  = counts + kN;
  int*   row_fill = row_ptr + (kN + 2);           // +1 entry, +1 pad for align
  int2*  edge_pairs = (int2*)(row_fill + kN);     // offset is 8B-aligned

  const int ne4_blocks  = (int)(kNE / 4 / 256);   // 9375
  const int edge_blocks = (kNNZ + 255) / 256;     // 9375
  const int n_blocks    = (kN + 255) / 256;       // 586
  const int row_blocks  = kN / 8;                 // 18750 (8 waves/block)

  init_concat<<<ne4_blocks, 256, 0, stream>>>(user_emb, item_emb, cur, acc);
  zero_i32<<<n_blocks, 256, 0, stream>>>(counts, kN);
  hist_rows<<<edge_blocks, 256, 0, stream>>>(rows, counts);
  scan_counts<<<1, 1024, 0, stream>>>(counts, row_ptr, row_fill, kN);
  scatter_edges<<<edge_blocks, 256, 0, stream>>>(rows, cols, vals, row_fill,
                                                 edge_pairs);
  // 3 LightGCN layers, ping-ponging cur/nxt.
  spmm_norm_acc<<<row_blocks, 256, 0, stream>>>(cur, nxt, acc, row_ptr,
                                                edge_pairs);
  spmm_norm_acc<<<row_blocks, 256, 0, stream>>>(nxt, cur, acc, row_ptr,
                                                edge_pairs);
  spmm_norm_acc<<<row_blocks, 256, 0, stream>>>(cur, nxt, acc, row_ptr,
                                                edge_pairs);
  finalize_mean<<<ne4_blocks, 256, 0, stream>>>(acc, out);
}